// MambaModel_65884798320658
// MI455X (gfx1250) — compile-verified
//
#include <hip/hip_runtime.h>
#include <hip/hip_bf16.h>

typedef __attribute__((ext_vector_type(2))) float v2f;
typedef __attribute__((ext_vector_type(8))) float v8f;

#define D_MODEL 256
#define D_INNER 512
#define SEQ_L   1024
#define BATCH   8
#define NSTATE  16
#define DT_RANK 16
#define NPROJ   48              // dt_rank + 2*N
#define NLAYERS 8
#define ROWS    (BATCH*SEQ_L)   // 8192

__device__ __forceinline__ float silu_f(float x) { return x / (1.f + __expf(-x)); }

// ---------------- encoder: h[b,l,m] = sum_c x[b,c,l]*ew[c,m] + eb[m] ----------------
__global__ void encoder_kernel(const float* __restrict__ x,
                               const float* __restrict__ ew,
                               const float* __restrict__ eb,
                               float* __restrict__ h)
{
    int m   = threadIdx.x;            // 0..255
    int row = blockIdx.x;             // b*L + l
    int b = row >> 10, l = row & 1023;
    const float* xp = x + (long)b * 3 * SEQ_L + l;
    float acc = eb[m];
    acc += xp[0]         * ew[0 * D_MODEL + m];
    acc += xp[SEQ_L]     * ew[1 * D_MODEL + m];
    acc += xp[2 * SEQ_L] * ew[2 * D_MODEL + m];
    h[(long)row * D_MODEL + m] = acc;
}

// ---------------- generic fp32 WMMA GEMM: C = act(A @ W + bias) ----------------
// A: M x K (row stride lda), W: K x N (row stride ldw), C: M x N (row stride ldc)
// act: 0 = identity, 1 = softplus
// Register-blocked: each wave computes an (MT*16) x (NT*16) macro-tile, reusing
// MT A-fragments against NT B-fragments -> MT*NT WMMAs per 4-wide K-step.
// 8 waves per block; all tile counts divisible by 8 so EXEC is all-ones.
template<int MT, int NT>
__global__ void wmma_gemm_f32(const float* __restrict__ A, int lda,
                              const float* __restrict__ W, int ldw,
                              const float* __restrict__ bias,
                              float* __restrict__ C, int ldc,
                              int M, int N, int K, int act)
{
    const int wave   = threadIdx.x >> 5;
    const int lane   = threadIdx.x & 31;
    const int tilesN = N / (NT * 16);
    const long tile  = (long)blockIdx.x * 8 + wave;
    const int  tmb   = (int)(tile / tilesN) * (MT * 16);
    const int  tnb   = (int)(tile % tilesN) * (NT * 16);
    if (tmb >= M) return;                     // wave-uniform guard

    const int rc = lane & 15;                 // A row / B col within tile
    const int kp = (lane >> 4) << 1;          // K-pair: 0 (lanes 0-15), 2 (lanes 16-31)

    const float* Ap[MT];
    #pragma unroll
    for (int i = 0; i < MT; ++i)
        Ap[i] = A + (long)(tmb + i * 16 + rc) * lda + kp;
    const float* Wp = W + (long)kp * ldw + tnb + rc;

    v8f acc[MT][NT] = {};

    for (int kk = 0; kk < K; kk += 4) {
        v2f a[MT], b[NT];
        #pragma unroll
        for (int i = 0; i < MT; ++i)
            a[i] = *(const v2f*)(Ap[i] + kk);            // A[M=rc][kp], [kp+1]
        #pragma unroll
        for (int j = 0; j < NT; ++j) {
            b[j].x = Wp[j * 16];                         // B[kp][N=rc]
            b[j].y = Wp[j * 16 + ldw];                   // B[kp+1][N=rc]
        }
        Wp += (long)4 * ldw;
        #pragma unroll
        for (int i = 0; i < MT; ++i)
            #pragma unroll
            for (int j = 0; j < NT; ++j)
                acc[i][j] = __builtin_amdgcn_wmma_f32_16x16x4_f32(
                                false, a[i], false, b[j],
                                (short)0, acc[i][j], false, false);
    }

    const int rhalf = (lane >> 4) << 3;       // +8 rows for upper half-wave
    #pragma unroll
    for (int j = 0; j < NT; ++j) {
        const int colg = tnb + j * 16 + rc;
        const float bv = bias ? bias[colg] : 0.f;
        #pragma unroll
        for (int i = 0; i < MT; ++i) {
            #pragma unroll
            for (int r = 0; r < 8; ++r) {
                float v = acc[i][j][r] + bv;
                if (act == 1) v = (v > 20.f) ? v : log1pf(__expf(v));   // softplus
                C[(long)(tmb + i * 16 + rhalf + r) * ldc + colg] = v;
            }
        }
    }
}

// ---------------- depthwise causal conv (K=4) + SiLU; u taken from xz[..., :512] ----------------
__global__ void conv_silu_kernel(const float* __restrict__ xz,
                                 const float* __restrict__ cw,   // (512,4)
                                 const float* __restrict__ cb,   // (512)
                                 float* __restrict__ u)          // (ROWS,512)
{
    long idx = (long)blockIdx.x * 256 + threadIdx.x;  // over ROWS*512
    int  d   = (int)(idx & 511);
    long row = idx >> 9;
    int  l   = (int)(row & 1023);
    long brow = row - l;                              // b*L
    float acc = cb[d];
    #pragma unroll
    for (int k = 0; k < 4; ++k) {
        int ls = l - 3 + k;
        if (ls >= 0) acc += xz[(brow + ls) * (2 * D_INNER) + d] * cw[d * 4 + k];
    }
    u[idx] = silu_f(acc);
}

// ---------------- selective scan + gating: ym = scan(u,dt,B,C,A,D) * silu(z) ----------------
// One lane per (b,d) channel; 16-float state held in VGPRs.
// B_t/C_t (32 shared floats) staged straight into LDS with the CDNA5 async
// memory->LDS path (GLOBAL_LOAD_ASYNC_TO_LDS_B32, ASYNCcnt), no VGPR bounce.
__global__ void scan_kernel(const float* __restrict__ u,     // (ROWS,512)
                            const float* __restrict__ dt,    // (ROWS,512)
                            const float* __restrict__ proj,  // (ROWS,48): B @16, C @32
                            const float* __restrict__ xz,    // (ROWS,1024): z @512
                            const float* __restrict__ A_log, // (512,16) layer slice
                            const float* __restrict__ Dv,    // (512)
                            float* __restrict__ ym)          // (ROWS,512)
{
    int b = blockIdx.x >> 1;
    int d = ((blockIdx.x & 1) << 8) + threadIdx.x;
    __shared__ float sBC[32];

    float Aa[NSTATE], s[NSTATE];
    #pragma unroll
    for (int n = 0; n < NSTATE; ++n) {
        Aa[n] = -__expf(A_log[d * NSTATE + n]);
        s[n]  = 0.f;
    }
    const float Dd  = Dv[d];
    const long base = (long)b * SEQ_L;

    // LDS byte offset of sBC for this lane (low 32 bits of the flat shared addr)
    unsigned ldsOff = (unsigned)(size_t)&sBC[threadIdx.x & 31];

    for (int t = 0; t < SEQ_L; ++t) {
        long row = base + t;
        if (threadIdx.x < 32) {               // wave 0 only (wave-uniform branch)
            unsigned long long gaddr =
                (unsigned long long)(proj + row * NPROJ + DT_RANK + threadIdx.x);
            asm volatile("global_load_async_to_lds_b32 %0, %1, off"
                         :: "v"(ldsOff), "v"(gaddr) : "memory");
            asm volatile("s_wait_asynccnt 0x0" ::: "memory");
        }
        __syncthreads();

        float dtv = dt[row * D_INNER + d];
        float uv  = u [row * D_INNER + d];
        float du  = dtv * uv;
        float y   = 0.f;
        #pragma unroll
        for (int n = 0; n < NSTATE; ++n) {
            s[n] = __expf(dtv * Aa[n]) * s[n] + du * sBC[n];
            y   += s[n] * sBC[NSTATE + n];
        }
        y += uv * Dd;
        float zv = xz[row * (2 * D_INNER) + D_INNER + d];
        ym[row * D_INNER + d] = y * silu_f(zv);
        __syncthreads();
    }
}

// ---------------- residual add + layernorm (in place into h) ----------------
__global__ void add_ln_kernel(const float* __restrict__ t,
                              float* __restrict__ h,
                              const float* __restrict__ g,
                              const float* __restrict__ bb)
{
    __shared__ float red[D_MODEL];
    long row = blockIdx.x;
    int  m   = threadIdx.x;
    float v = t[row * D_MODEL + m] + h[row * D_MODEL + m];

    red[m] = v; __syncthreads();
    for (int stp = D_MODEL / 2; stp > 0; stp >>= 1) {
        if (m < stp) red[m] += red[m + stp];
        __syncthreads();
    }
    float mean = red[0] * (1.f / D_MODEL);
    __syncthreads();

    float dv = v - mean;
    red[m] = dv * dv; __syncthreads();
    for (int stp = D_MODEL / 2; stp > 0; stp >>= 1) {
        if (m < stp) red[m] += red[m + stp];
        __syncthreads();
    }
    float var = red[0] * (1.f / D_MODEL);
    h[row * D_MODEL + m] = dv * rsqrtf(var + 1e-5f) * g[m] + bb[m];
}

// ---------------- mean-pool over L ----------------
__global__ void pool_kernel(const float* __restrict__ h, float* __restrict__ pooled)
{
    int b = blockIdx.x, m = threadIdx.x;
    float acc = 0.f;
    for (int l = 0; l < SEQ_L; ++l)
        acc += h[((long)b * SEQ_L + l) * D_MODEL + m];
    pooled[b * D_MODEL + m] = acc * (1.f / SEQ_L);
}

// ---------------- decoder: out[b,o] = pooled[b]@dec_w + dec_b ----------------
__global__ void decode_kernel(const float* __restrict__ pooled,
                              const float* __restrict__ dw,
                              const float* __restrict__ db,
                              float* __restrict__ out)
{
    int t = threadIdx.x;
    if (t >= BATCH * 10) return;
    int b = t / 10, o = t % 10;
    float acc = db[o];
    for (int m = 0; m < D_MODEL; ++m)
        acc += pooled[b * D_MODEL + m] * dw[m * 10 + o];
    out[b * 10 + o] = acc;
}

extern "C" void kernel_launch(void* const* d_in, const int* in_sizes, int n_in,
                              void* d_out, int out_size, void* d_ws, size_t ws_size,
                              hipStream_t stream)
{
    (void)in_sizes; (void)n_in; (void)out_size; (void)ws_size;

    const float* x      = (const float*)d_in[0];
    const float* enc_w  = (const float*)d_in[1];
    const float* enc_b  = (const float*)d_in[2];
    const float* ipw    = (const float*)d_in[3];   // (8,256,1024)
    const float* cw     = (const float*)d_in[4];   // (8,512,4)
    const float* cb     = (const float*)d_in[5];   // (8,512)
    const float* xpw    = (const float*)d_in[6];   // (8,512,48)
    const float* dtw    = (const float*)d_in[7];   // (8,16,512)
    const float* dtb    = (const float*)d_in[8];   // (8,512)
    const float* alog   = (const float*)d_in[9];   // (8,512,16)
    const float* Dp     = (const float*)d_in[10];  // (8,512)
    const float* opw    = (const float*)d_in[11];  // (8,512,256)
    const float* lng    = (const float*)d_in[12];  // (8,256)
    const float* lnb    = (const float*)d_in[13];  // (8,256)
    const float* dec_w  = (const float*)d_in[14];  // (256,10)
    const float* dec_b  = (const float*)d_in[15];  // (10)
    float* out = (float*)d_out;

    // workspace layout (floats)
    float* ws = (float*)d_ws;
    float* h    = ws;                               // 8192*256
    float* xz   = h    + (long)ROWS * D_MODEL;      // 8192*1024
    float* u    = xz   + (long)ROWS * 2 * D_INNER;  // 8192*512
    float* proj = u    + (long)ROWS * D_INNER;      // 8192*48
    float* dt   = proj + (long)ROWS * NPROJ;        // 8192*512
    float* ym   = dt   + (long)ROWS * D_INNER;      // 8192*512
    float* tmp  = ym   + (long)ROWS * D_INNER;      // 8192*256
    float* pool = tmp  + (long)ROWS * D_MODEL;      // 8*256

    // encoder
    encoder_kernel<<<dim3(ROWS), dim3(D_MODEL), 0, stream>>>(x, enc_w, enc_b, h);

    for (int i = 0; i < NLAYERS; ++i) {
        const float* ipw_i  = ipw  + (long)i * D_MODEL * 2 * D_INNER;
        const float* cw_i   = cw   + (long)i * D_INNER * 4;
        const float* cb_i   = cb   + (long)i * D_INNER;
        const float* xpw_i  = xpw  + (long)i * D_INNER * NPROJ;
        const float* dtw_i  = dtw  + (long)i * DT_RANK * D_INNER;
        const float* dtb_i  = dtb  + (long)i * D_INNER;
        const float* alog_i = alog + (long)i * D_INNER * NSTATE;
        const float* D_i    = Dp   + (long)i * D_INNER;
        const float* opw_i  = opw  + (long)i * D_INNER * D_MODEL;
        const float* lng_i  = lng  + (long)i * D_MODEL;
        const float* lnb_i  = lnb  + (long)i * D_MODEL;

        // xz = h @ in_proj   (8192x256 @ 256x1024), 32x64 macro-tiles
        {
            int blocks = (ROWS / 32) * ((2 * D_INNER) / 64) / 8;   // 512
            wmma_gemm_f32<2, 4><<<dim3(blocks), dim3(256), 0, stream>>>(
                h, D_MODEL, ipw_i, 2 * D_INNER, nullptr,
                xz, 2 * D_INNER, ROWS, 2 * D_INNER, D_MODEL, 0);
        }

        // u = silu(causal_conv(xz[..., :512]))
        conv_silu_kernel<<<dim3((ROWS * D_INNER) / 256), dim3(256), 0, stream>>>(
            xz, cw_i, cb_i, u);

        // proj = u @ x_proj  (8192x512 @ 512x48), 32x48 macro-tiles
        {
            int blocks = (ROWS / 32) * (NPROJ / 48) / 8;            // 32
            wmma_gemm_f32<2, 3><<<dim3(blocks), dim3(256), 0, stream>>>(
                u, D_INNER, xpw_i, NPROJ, nullptr,
                proj, NPROJ, ROWS, NPROJ, D_INNER, 0);
        }

        // dt = softplus(proj[..., :16] @ dt_w + dt_b)  (8192x16 @ 16x512)
        {
            int blocks = (ROWS / 32) * (D_INNER / 64) / 8;          // 256
            wmma_gemm_f32<2, 4><<<dim3(blocks), dim3(256), 0, stream>>>(
                proj, NPROJ, dtw_i, D_INNER, dtb_i,
                dt, D_INNER, ROWS, D_INNER, DT_RANK, 1);
        }

        // selective scan + gating -> ym
        scan_kernel<<<dim3(BATCH * 2), dim3(256), 0, stream>>>(
            u, dt, proj, xz, alog_i, D_i, ym);

        // tmp = ym @ out_proj  (8192x512 @ 512x256)
        {
            int blocks = (ROWS / 32) * (D_MODEL / 64) / 8;          // 128
            wmma_gemm_f32<2, 4><<<dim3(blocks), dim3(256), 0, stream>>>(
                ym, D_INNER, opw_i, D_MODEL, nullptr,
                tmp, D_MODEL, ROWS, D_MODEL, D_INNER, 0);
        }

        // h = layernorm(tmp + h)
        add_ln_kernel<<<dim3(ROWS), dim3(D_MODEL), 0, stream>>>(tmp, h, lng_i, lnb_i);
    }

    pool_kernel<<<dim3(BATCH), dim3(D_MODEL), 0, stream>>>(h, pool);
    decode_kernel<<<dim3(1), dim3(128), 0, stream>>>(pool, dec_w, dec_b, out);
}